// Decoder_23639499997380
// MI455X (gfx1250) — compile-verified
//
#include <hip/hip_runtime.h>
#include <hip/hip_bf16.h>

// ---------------------------------------------------------------------------
// 2-layer GCN for MI455X (gfx1250, wave32).
//   xw  = X @ W                    (fp32 WMMA v_wmma_f32_16x16x4_f32)
//   h   = D^-1/2 (A+I) D^-1/2 xw   (self-loop init + per-edge atomic scatter)
//   act = relu(h+b1) / tanh(h+b2)
// Aggregation is the memory-bound phase (~2.5 GB; scatter destinations h1
// [51MB] and out [25.6MB] fit the 192MB L2, so fp32 atomics stay L2-resident).
// GEMMs (~5 GFLOP fp32) use native fp32 WMMA to match fp32 reference numerics.
// Scatter uses unsafeAtomicAdd -> native global_atomic_add_f32 (no-return,
// STOREcnt-tracked) instead of a CAS retry loop.
// ---------------------------------------------------------------------------

typedef float v2f __attribute__((ext_vector_type(2)));
typedef float v8f __attribute__((ext_vector_type(8)));

// Native fp32 atomic add (global_atomic_add_f32, no return value needed).
__device__ __forceinline__ void atomic_fadd(float* p, float v) {
    unsafeAtomicAdd(p, v);
}

// ---------------- degree / normalization ----------------------------------

__global__ void k_deg_init(float* __restrict__ deg, int n) {
    int i = blockIdx.x * blockDim.x + threadIdx.x;
    if (i < n) deg[i] = 1.0f;  // self-loop contributes 1 to every destination
}

__global__ void k_deg_accum(const long long* __restrict__ col,
                            float* __restrict__ deg, int e) {
    int i = blockIdx.x * blockDim.x + threadIdx.x;
    if (i < e) atomic_fadd(&deg[col[i]], 1.0f);
}

__global__ void k_deg_rsqrt(float* __restrict__ deg, int n) {
    int i = blockIdx.x * blockDim.x + threadIdx.x;
    if (i < n) deg[i] = rsqrtf(deg[i]);  // deg >= 1 always (self-loops)
}

// ---------------- fp32 WMMA GEMM: C[M x NCOLS] = A[M x 128] * B[128 x NCOLS]
// One wave computes one 16x16 tile of C; K=128 = 32 steps of 16x16x4.
// A-frag layout (32-bit A 16x4): lanes 0-15 -> M=lane, K = k0 + {0,1};
//                                lanes 16-31 -> M=lane-16, K = k0 + {2,3}.
// B-frag (4x16, rows striped across lanes within a VGPR):
//   v0: lanes 0-15 = row k0,   lanes 16-31 = row k0+2 (N = lane&15)
//   v1: lanes 0-15 = row k0+1, lanes 16-31 = row k0+3
// C/D layout: VGPR r, lanes 0-15 -> M=r, lanes 16-31 -> M=r+8.

template <int NCOLS>
__global__ void k_gemm_wmma(const float* __restrict__ A,
                            const float* __restrict__ B,
                            float* __restrict__ C, int mTiles) {
    const int nTiles = NCOLS / 16;
    int wid  = threadIdx.x >> 5;
    int tile = blockIdx.x * (blockDim.x >> 5) + wid;
    if (tile >= mTiles * nTiles) return;   // wave-uniform: EXEC stays all-ones
    int tM = tile / nTiles;
    int tN = tile - tM * nTiles;

    int lane = threadIdx.x & 31;
    int l    = lane & 15;
    int hi   = lane >> 4;                  // 0 or 1 (upper half-wave)

    const float* aRow = A + (size_t)(tM * 16 + l) * 128;
    const float* bCol = B + tN * 16 + l;

    v8f acc = {};
#pragma unroll 4
    for (int k0 = 0; k0 < 128; k0 += 4) {
        int ka = k0 + 2 * hi;
        v2f a = *(const v2f*)(aRow + ka);      // K=ka, ka+1  (8B aligned)
        v2f b;
        b.x = bCol[(size_t)(ka + 0) * NCOLS];  // row ka
        b.y = bCol[(size_t)(ka + 1) * NCOLS];  // row ka+1
        acc = __builtin_amdgcn_wmma_f32_16x16x4_f32(
            /*neg_a=*/false, a, /*neg_b=*/false, b,
            /*c_mod=*/(short)0, acc, /*reuse_a=*/false, /*reuse_b=*/false);
    }

    float* cBase = C + (size_t)(tM * 16 + hi * 8) * NCOLS + tN * 16 + l;
#pragma unroll
    for (int r = 0; r < 8; ++r)
        cBase[(size_t)r * NCOLS] = acc[r];
}

// Scalar tail for M % 16 != 0 (not hit for N=100000, kept for generality).
template <int NCOLS>
__global__ void k_gemm_tail(const float* __restrict__ A,
                            const float* __restrict__ B,
                            float* __restrict__ C, int rowStart, int M) {
    int r = rowStart + blockIdx.x;
    int c = threadIdx.x;
    if (r >= M || c >= NCOLS) return;
    float acc = 0.f;
    for (int k = 0; k < 128; ++k)
        acc += A[(size_t)r * 128 + k] * B[(size_t)k * NCOLS + c];
    C[(size_t)r * NCOLS + c] = acc;
}

// ---------------- aggregation ----------------------------------------------

// out[i,:] = xw[i,:] * dinv[i]^2   (self-loop term; also initializes out)
__global__ void k_agg_self(const float* __restrict__ xw,
                           const float* __restrict__ dinv,
                           float* __restrict__ out, long long total, int shift) {
    long long i = (long long)blockIdx.x * blockDim.x + threadIdx.x;
    if (i >= total) return;
    float d = dinv[i >> shift];
    out[i] = xw[i] * d * d;
}

// One wave per edge: gather xw[row,:], scale by dinv[row]*dinv[col],
// scatter-add into out[col,:]. C/32 contiguous floats per lane.
template <int C>
__global__ void k_agg_edges(const long long* __restrict__ row,
                            const long long* __restrict__ col,
                            const float* __restrict__ xw,
                            const float* __restrict__ dinv,
                            float* __restrict__ out, int e) {
    int ei = blockIdx.x * (blockDim.x >> 5) + (threadIdx.x >> 5);
    if (ei >= e) return;
    int lane = threadIdx.x & 31;
    long long r = row[ei];
    long long c = col[ei];
    float norm = dinv[r] * dinv[c];

    constexpr int V = C / 32;  // 4 (C=128) or 2 (C=64)
    const float* src = xw + r * (long long)C + lane * V;
    float*       dst = out + c * (long long)C + lane * V;
#pragma unroll
    for (int j = 0; j < V; ++j)
        atomic_fadd(dst + j, src[j] * norm);
}

// h[i] = act(h[i] + bias[i & (c-1)])   act: 0 = relu, 1 = tanh
template <int ACT>
__global__ void k_bias_act(float* __restrict__ h,
                           const float* __restrict__ bias,
                           long long total, int cMask) {
    long long i = (long long)blockIdx.x * blockDim.x + threadIdx.x;
    if (i >= total) return;
    float v = h[i] + bias[i & cMask];
    h[i] = (ACT == 0) ? fmaxf(v, 0.0f) : tanhf(v);
}

// ---------------- host driver ----------------------------------------------

extern "C" void kernel_launch(void* const* d_in, const int* in_sizes, int n_in,
                              void* d_out, int out_size, void* d_ws, size_t ws_size,
                              hipStream_t stream) {
    const float*     z   = (const float*)d_in[0];
    const long long* ei  = (const long long*)d_in[1];  // int64 [2, E] flat
    const float*     W1  = (const float*)d_in[2];
    const float*     b1  = (const float*)d_in[3];
    const float*     W2  = (const float*)d_in[4];
    const float*     b2  = (const float*)d_in[5];

    const int n = in_sizes[0] / 128;   // 100000 nodes
    const int e = in_sizes[1] / 2;     // 1.6M edges
    const long long* rowIdx = ei;      // source nodes
    const long long* colIdx = ei + e;  // destination nodes

    // Workspace layout (fp32): dinv[n] | xw1[n*128] | h1[n*128] | xw2[n*64]
    float* dinv = (float*)d_ws;
    float* xw1  = dinv + n;
    float* h1   = xw1 + (size_t)n * 128;
    float* xw2  = h1  + (size_t)n * 128;
    float* out  = (float*)d_out;

    const int T = 256;

    // --- symmetric normalization coefficients ------------------------------
    k_deg_init <<<(n + T - 1) / T, T, 0, stream>>>(dinv, n);
    k_deg_accum<<<(e + T - 1) / T, T, 0, stream>>>(colIdx, dinv, e);
    k_deg_rsqrt<<<(n + T - 1) / T, T, 0, stream>>>(dinv, n);

    const int mTiles = n / 16;
    const int mRem   = n - mTiles * 16;

    // --- layer 1: xw1 = z @ W1 (WMMA), aggregate, relu(+b1) ----------------
    {
        int tiles = mTiles * (128 / 16);
        k_gemm_wmma<128><<<(tiles + 3) / 4, 128, 0, stream>>>(z, W1, xw1, mTiles);
        if (mRem > 0)
            k_gemm_tail<128><<<mRem, 128, 0, stream>>>(z, W1, xw1, mTiles * 16, n);
    }
    long long tot1 = (long long)n * 128;
    k_agg_self<<<(int)((tot1 + T - 1) / T), T, 0, stream>>>(xw1, dinv, h1, tot1, 7);
    k_agg_edges<128><<<(e + 7) / 8, T, 0, stream>>>(rowIdx, colIdx, xw1, dinv, h1, e);
    k_bias_act<0><<<(int)((tot1 + T - 1) / T), T, 0, stream>>>(h1, b1, tot1, 127);

    // --- layer 2: xw2 = h1 @ W2 (WMMA), aggregate, tanh(+b2) ---------------
    {
        int tiles = mTiles * (64 / 16);
        k_gemm_wmma<64><<<(tiles + 3) / 4, 128, 0, stream>>>(h1, W2, xw2, mTiles);
        if (mRem > 0)
            k_gemm_tail<64><<<mRem, 64, 0, stream>>>(h1, W2, xw2, mTiles * 16, n);
    }
    long long tot2 = (long long)n * 64;
    k_agg_self<<<(int)((tot2 + T - 1) / T), T, 0, stream>>>(xw2, dinv, out, tot2, 6);
    k_agg_edges<64><<<(e + 7) / 8, T, 0, stream>>>(rowIdx, colIdx, xw2, dinv, out, e);
    k_bias_act<1><<<(int)((tot2 + T - 1) / T), T, 0, stream>>>(out, b2, tot2, 63);
}